// RoIHeads_24575802867948
// MI455X (gfx1250) — compile-verified
//
#include <hip/hip_runtime.h>
#include <hip/hip_bf16.h>
#include <math.h>

// ---------------- problem constants (from reference) ----------------
#define B_    4
#define N_    4096
#define G_    128
#define C_    16
#define KNMS_ 512
#define DETS_ 100
#define BATCH_PER_IMG_ 512
#define POS_PER_IMG_   128

#define SCORE_T_ 0.05f
#define NMS_T_   0.1f
#define MINSZ_   0.01f
#define FG_T_    0.5f
#define CLS_OFF_ 4096.0f
#define PI_F     3.14159265358979323846f
#define CLIP_V   4.135166556742356f   /* log(1000/16) */
#define EPS_IOU  1e-8f
#define EPS_IN   1e-6f

typedef float v2f __attribute__((ext_vector_type(2)));
typedef float v8f __attribute__((ext_vector_type(8)));
typedef unsigned int u32x4 __attribute__((ext_vector_type(4)));
typedef int i32x4v __attribute__((ext_vector_type(4)));
typedef int i32x8v __attribute__((ext_vector_type(8)));

// ---------------- WMMA distance Gram matrix --------------------------
// D[m][n] = A[m][:] dot B[:][n] with A-row = [x, y, |c|^2, 1] and
// B-col = [-2x, -2y, 1, |c|^2]  =>  D[m][n] = |c_m - c_n|^2  (exact f32)
__device__ __forceinline__ v8f wmma_dist(v2f a, v2f b, v8f c) {
#if defined(__AMDGCN__)
  return __builtin_amdgcn_wmma_f32_16x16x4_f32(false, a, false, b, (short)0, c,
                                               false, false);
#else
  (void)a; (void)b;
  return c;
#endif
}

// ---------------- Tensor Data Mover: 1-D global -> LDS DMA -----------
// Builds a D# (CDNA5 ISA) for a contiguous copy of
// `ndwords` dwords and issues TENSOR_LOAD_TO_LDS; waits TENSORcnt==0.
// Caller must be a single wave; follow with a workgroup barrier.
__device__ __forceinline__ void tdm_load_1d(const void* gaddr,
                                            unsigned lds_off,
                                            unsigned ndwords) {
#if defined(__AMDGCN__)
  unsigned long long ga = (unsigned long long)(size_t)gaddr;
  u32x4 g0;
  g0[0] = 1u;                                    // count=1 (valid user D#)
  g0[1] = lds_off;                               // lds_addr (bytes)
  g0[2] = (unsigned)(ga & 0xffffffffu);          // global_addr[31:0]
  g0[3] = (unsigned)((ga >> 32) & 0x01ffffffu)   // global_addr[56:32]
          | 0x80000000u;                         // type=2 ("image")
  i32x8v g1;
  g1[0] = 0x00020000;                            // mask=0, data_size=2 (4B)
  g1[1] = (int)((ndwords & 0xffffu) << 16);      // tensor_dim0[15:0] @63:48
  g1[2] = (int)(((ndwords >> 16) & 0xffffu)      // tensor_dim0[31:16] @79:64
                | (1u << 16));                   // tensor_dim1=1 @95:80
  g1[3] = (int)((ndwords & 0xffffu) << 16);      // tile_dim0 @127:112
  g1[4] = 1;                                     // tile_dim1=1 @143:128
  g1[5] = (int)ndwords;                          // dim0_stride[31:0] @191:160
  g1[6] = 0;                                     // dim0_stride[47:32], d1strd
  g1[7] = 0;
  i32x4v z4 = {0, 0, 0, 0};
#if __clang_major__ >= 23
  i32x8v z8 = {0, 0, 0, 0, 0, 0, 0, 0};
  __builtin_amdgcn_tensor_load_to_lds(g0, g1, z4, z4, z8, 0);
#else
  __builtin_amdgcn_tensor_load_to_lds(g0, g1, z4, z4, 0);
#endif
  __builtin_amdgcn_s_wait_tensorcnt((short)0);
#else
  (void)gaddr; (void)lds_off; (void)ndwords;
#endif
}

// LDS byte offset of a __shared__ object: LDS aperture occupies flat-address
// bits [63:32], so the low 32 bits are the in-LDS offset.
__device__ __forceinline__ unsigned lds_offset_of(const void* p) {
  return (unsigned)(size_t)p;
}

// ---------------- rotated-box geometry (reference-faithful) ----------
struct RBox { float cx, cy, w, h, a; };

__device__ __forceinline__ float crossf(float ax, float ay, float bx, float by) {
  return ax * by - ay * bx;
}

__device__ __forceinline__ void rbox_corners(const RBox& b, float* px, float* py) {
  float c = cosf(b.a), s = sinf(b.a);
  const float lx[4] = {0.5f, 0.5f, -0.5f, -0.5f};
  const float ly[4] = {0.5f, -0.5f, -0.5f, 0.5f};
#pragma unroll
  for (int i = 0; i < 4; ++i) {
    float dx = lx[i] * b.w, dy = ly[i] * b.h;
    px[i] = b.cx + dx * c - dy * s;
    py[i] = b.cy + dx * s + dy * c;
  }
}

__device__ __forceinline__ bool poly_inside(float ptx, float pty,
                                            const float* qx, const float* qy) {
  bool allp = true, alln = true;
#pragma unroll
  for (int e = 0; e < 4; ++e) {
    int e2 = (e + 1) & 3;
    float d = crossf(qx[e2] - qx[e], qy[e2] - qy[e], ptx - qx[e], pty - qy[e]);
    allp = allp && (d >= -EPS_IN);
    alln = alln && (d <= EPS_IN);
  }
  return allp || alln;
}

__device__ float rot_inter_area(const float* ax, const float* ay,
                                const float* bx, const float* by) {
  float px[24], py[24], ang[24];
  bool msk[24];
#pragma unroll
  for (int i = 0; i < 4; ++i) {
    int i2 = (i + 1) & 3;
    float rx = ax[i2] - ax[i], ry = ay[i2] - ay[i];
#pragma unroll
    for (int j = 0; j < 4; ++j) {
      int j2 = (j + 1) & 3;
      float sx = bx[j2] - bx[j], sy = by[j2] - by[j];
      float den = crossf(rx, ry, sx, sy);
      float qpx = bx[j] - ax[i], qpy = by[j] - ay[i];
      float dsafe = (fabsf(den) < EPS_IOU) ? 1.0f : den;
      float t = crossf(qpx, qpy, sx, sy) / dsafe;
      float u = crossf(qpx, qpy, rx, ry) / dsafe;
      bool ok = (fabsf(den) >= EPS_IOU) && (t >= 0.0f) && (t <= 1.0f) &&
                (u >= 0.0f) && (u <= 1.0f);
      int k = i * 4 + j;
      px[k] = ax[i] + t * rx;
      py[k] = ay[i] + t * ry;
      msk[k] = ok;
    }
  }
#pragma unroll
  for (int i = 0; i < 4; ++i) {
    px[16 + i] = ax[i]; py[16 + i] = ay[i];
    msk[16 + i] = poly_inside(ax[i], ay[i], bx, by);
    px[20 + i] = bx[i]; py[20 + i] = by[i];
    msk[20 + i] = poly_inside(bx[i], by[i], ax, ay);
  }
  int cnt = 0;
  float sx = 0.0f, sy = 0.0f;
#pragma unroll
  for (int k = 0; k < 24; ++k) {
    if (msk[k]) { cnt++; sx += px[k]; sy += py[k]; }
  }
  float inv = 1.0f / (float)((cnt > 1) ? cnt : 1);
  float cx = sx * inv, cy = sy * inv;
#pragma unroll
  for (int k = 0; k < 24; ++k) {
    ang[k] = msk[k] ? atan2f(py[k] - cy, px[k] - cx) : 1e9f;
  }
  // odd-even transposition sort network (static indices -> stays in VGPRs)
#pragma unroll
  for (int p = 0; p < 24; ++p) {
#pragma unroll
    for (int i = (p & 1); i < 23; i += 2) {
      if (ang[i] > ang[i + 1]) {
        float t0;
        t0 = ang[i]; ang[i] = ang[i + 1]; ang[i + 1] = t0;
        t0 = px[i];  px[i]  = px[i + 1];  px[i + 1]  = t0;
        t0 = py[i];  py[i]  = py[i + 1];  py[i + 1]  = t0;
      }
    }
  }
  float fx = px[0], fy = py[0];
#pragma unroll
  for (int k = 0; k < 24; ++k) {
    bool mk = ang[k] < 1e8f;
    px[k] = mk ? px[k] : fx;
    py[k] = mk ? py[k] : fy;
  }
  float a2 = 0.0f;
#pragma unroll
  for (int k = 0; k < 24; ++k) {
    int k2 = (k + 1) % 24;
    a2 += px[k] * py[k2] - py[k] * px[k2];
  }
  float area = 0.5f * fabsf(a2);
  return (cnt >= 3) ? area : 0.0f;
}

__device__ float rotated_iou_boxes(const RBox& b1, const RBox& b2) {
  float ax[4], ay[4], bx[4], by[4];
  rbox_corners(b1, ax, ay);
  rbox_corners(b2, bx, by);
  float inter = rot_inter_area(ax, ay, bx, by);
  float a1 = fabsf(b1.w * b1.h);
  float a2 = fabsf(b2.w * b2.h);
  return inter / (a1 + a2 - inter + EPS_IOU);
}

// ---------------- kernel 1: GT <-> proposal matching -----------------
// One wave per 16-proposal tile; GT table staged into LDS by the TDM.
// WMMA produces the 16(gt) x 16(prop) squared-distance tile; exact IoU
// runs only where the enclosing circles overlap.
__global__ __launch_bounds__(128) void roih_match_kernel(
    const float* __restrict__ props, const float* __restrict__ gt,
    const int* __restrict__ gtlab, int* __restrict__ mlab_i,
    float* __restrict__ mlab_f) {
  __shared__ float sGT[G_ * 5];

  int wave = (int)((blockIdx.x * blockDim.x + threadIdx.x) >> 5);
  int lane = (int)(threadIdx.x & 31);
  int b = wave / (N_ / 16);       // block-uniform: 64 blocks per batch
  int pt = wave % (N_ / 16);
  int n = pt * 16 + (lane & 15);

  // TDM DMA: gt slice for this batch (128 boxes * 5 f32 = 640 dwords)
  if (threadIdx.x < 32) {
    tdm_load_1d(gt + (size_t)b * G_ * 5, lds_offset_of(sGT), G_ * 5);
  }
  __syncthreads();

  const float* pb = props + ((size_t)b * N_ + n) * 5;
  RBox P; P.cx = pb[0]; P.cy = pb[1]; P.w = pb[2]; P.h = pb[3]; P.a = pb[4];
  float rp = 0.5f * sqrtf(P.w * P.w + P.h * P.h);

  v2f Bop;
  if (lane < 16) { Bop[0] = -2.0f * P.cx; Bop[1] = -2.0f * P.cy; }
  else           { Bop[0] = 1.0f;         Bop[1] = P.cx * P.cx + P.cy * P.cy; }

  float best = -1.0f;
  int bestIdx = 0;
  int h = lane >> 4;

  for (int t = 0; t < G_ / 16; ++t) {
    int m = t * 16 + (lane & 15);
    float gx = sGT[m * 5 + 0], gy = sGT[m * 5 + 1];
    v2f Aop;
    if (lane < 16) { Aop[0] = gx;                Aop[1] = gy;   }
    else           { Aop[0] = gx * gx + gy * gy; Aop[1] = 1.0f; }

    v8f acc = {};
    acc = wmma_dist(Aop, Bop, acc);   // acc[v] = d^2(gt[t*16+v+8h], prop n)

#pragma unroll
    for (int v = 0; v < 8; ++v) {
      int g = t * 16 + v + 8 * h;
      RBox Gb; Gb.cx = sGT[g * 5 + 0]; Gb.cy = sGT[g * 5 + 1];
      Gb.w = sGT[g * 5 + 2]; Gb.h = sGT[g * 5 + 3]; Gb.a = sGT[g * 5 + 4];
      float rg = 0.5f * sqrtf(Gb.w * Gb.w + Gb.h * Gb.h);
      float rr = rp + rg;
      float iou = 0.0f;
      if (acc[v] < rr * rr) iou = rotated_iou_boxes(Gb, P);
      if (iou > best) { best = iou; bestIdx = g; }
      else if (iou == best && g < bestIdx) { bestIdx = g; }
    }
  }
  // combine the two lane-halves (gt rows v and v+8)
  float obest = __shfl_xor(best, 16, 32);
  int oidx = __shfl_xor(bestIdx, 16, 32);
  if (obest > best || (obest == best && oidx < bestIdx)) {
    best = obest; bestIdx = oidx;
  }
  if (lane < 16) {
    int lab = gtlab[b * G_ + bestIdx];
    int outl = (best >= FG_T_) ? lab : 0;   // FG_T == BG_T == 0.5
    mlab_i[b * N_ + n] = outl;
    mlab_f[b * N_ + n] = (float)outl;
  }
}

// ---------------- kernel 2: decode + softmax + candidate scores ------
__global__ __launch_bounds__(256) void roih_decode_kernel(
    const float* __restrict__ props, const float* __restrict__ logits,
    const float* __restrict__ reg, float* __restrict__ pred_boxes,
    float* __restrict__ cand) {
  int idx = (int)(blockIdx.x * blockDim.x + threadIdx.x);
  if (idx >= B_ * N_) return;
  int b = idx / N_;
  int n = idx % N_;
  const float* pb = props + (size_t)idx * 5;
  float px = pb[0], py = pb[1], pw = pb[2], ph = pb[3], pa = pb[4];

  const float* lg = logits + (size_t)idx * C_;
  float mx = -1e30f;
#pragma unroll
  for (int c = 0; c < C_; ++c) mx = fmaxf(mx, lg[c]);
  float e[C_];
  float sum = 0.0f;
#pragma unroll
  for (int c = 0; c < C_; ++c) { e[c] = __expf(lg[c] - mx); sum += e[c]; }
  float inv = 1.0f / sum;

  const float* rg = reg + (size_t)idx * C_ * 5;
  float* ob = pred_boxes + (size_t)idx * C_ * 5;
#pragma unroll
  for (int c = 0; c < C_; ++c) {
    float dx = rg[c * 5 + 0], dy = rg[c * 5 + 1];
    float dw = fminf(rg[c * 5 + 2], CLIP_V);
    float dh = fminf(rg[c * 5 + 3], CLIP_V);
    float da = rg[c * 5 + 4];
    float cx = dx * pw + px;
    float cy = dy * ph + py;
    float w = pw * __expf(dw);
    float hh = ph * __expf(dh);
    float x = pa + da + PI_F;
    x -= floorf(x * (1.0f / (2.0f * PI_F))) * (2.0f * PI_F);
    float a = x - PI_F;
    ob[c * 5 + 0] = cx; ob[c * 5 + 1] = cy; ob[c * 5 + 2] = w;
    ob[c * 5 + 3] = hh; ob[c * 5 + 4] = a;
    if (c >= 1) {
      float sc = e[c] * inv;
      bool valid = (sc > SCORE_T_) && (w > MINSZ_) && (hh > MINSZ_);
      cand[(size_t)b * (N_ * (C_ - 1)) + (size_t)n * (C_ - 1) + (c - 1)] =
          valid ? sc : -1.0f;
    }
  }
}

// ---------------- kernel 3: balanced subsampling ---------------------
__global__ __launch_bounds__(256) void roih_subsample_kernel(
    const int* __restrict__ mlab, const float* __restrict__ rnd,
    float* __restrict__ maskout) {
  __shared__ float sr[N_];
  __shared__ signed char sl[N_];
  __shared__ int cpos, cneg;
  int b = (int)blockIdx.x;
  if (threadIdx.x == 0) { cpos = 0; cneg = 0; }
  __syncthreads();
  for (int i = (int)threadIdx.x; i < N_; i += 256) {
    int l = mlab[b * N_ + i];
    sr[i] = rnd[b * N_ + i];
    signed char cls = (l > 0) ? (signed char)1 : (signed char)0;
    sl[i] = cls;
    if (cls) atomicAdd(&cpos, 1); else atomicAdd(&cneg, 1);
  }
  __syncthreads();
  int npos = min(cpos, POS_PER_IMG_);
  int nneg = min(cneg, BATCH_PER_IMG_ - npos);
  for (int i = (int)threadIdx.x; i < N_; i += 256) {
    signed char cls = sl[i];
    float ri = sr[i];
    int rank = 0;
    for (int j = 0; j < N_; ++j) {
      if (sl[j] == cls) {
        float rj = sr[j];
        rank += (int)((rj > ri) || (rj == ri && j < i));
      }
    }
    bool sel = cls ? (rank < npos) : (rank < nneg);
    maskout[b * N_ + i] = sel ? 1.0f : 0.0f;
  }
}

// ---------------- kernel 4: iterative top-512 per batch --------------
__global__ __launch_bounds__(512) void roih_topk_kernel(
    float* __restrict__ cand, float* __restrict__ tvals,
    int* __restrict__ tidx) {
  __shared__ float sv[512];
  __shared__ int si[512];
  int b = (int)blockIdx.x;
  float* c = cand + (size_t)b * (N_ * (C_ - 1));
  const int M = N_ * (C_ - 1);
  for (int it = 0; it < KNMS_; ++it) {
    float bv = -1e30f;
    int bi = 0;
    for (int i = (int)threadIdx.x; i < M; i += 512) {
      __builtin_prefetch(c + i + 512, 0, 0);   // global_prefetch_b8
      float v = c[i];
      if (v > bv) { bv = v; bi = i; }
    }
    sv[threadIdx.x] = bv;
    si[threadIdx.x] = bi;
    __syncthreads();
    for (int off = 256; off > 0; off >>= 1) {
      if ((int)threadIdx.x < off) {
        float v2 = sv[threadIdx.x + off];
        int i2 = si[threadIdx.x + off];
        if (v2 > sv[threadIdx.x] ||
            (v2 == sv[threadIdx.x] && i2 < si[threadIdx.x])) {
          sv[threadIdx.x] = v2;
          si[threadIdx.x] = i2;
        }
      }
      __syncthreads();
    }
    if (threadIdx.x == 0) {
      tvals[b * KNMS_ + it] = sv[0];
      tidx[b * KNMS_ + it] = si[0];
      c[si[0]] = -1e30f;   // remove from further rounds
    }
    __syncthreads();
  }
}

// ---------------- kernel 5: gather selected boxes --------------------
__global__ __launch_bounds__(256) void roih_gather_kernel(
    const float* __restrict__ pred_boxes, const int* __restrict__ tidx,
    float* __restrict__ selbox, float* __restrict__ selcx0,
    int* __restrict__ sellab) {
  int i = (int)(blockIdx.x * blockDim.x + threadIdx.x);
  if (i >= B_ * KNMS_) return;
  int b = i / KNMS_;
  int id = tidx[i];
  int n = id / (C_ - 1);
  int c = id % (C_ - 1) + 1;
  const float* sb = pred_boxes + (((size_t)b * N_ + n) * C_ + c) * 5;
  selbox[i * 5 + 0] = sb[0] + (float)c * CLS_OFF_;  // class-offset for NMS
  selbox[i * 5 + 1] = sb[1];
  selbox[i * 5 + 2] = sb[2];
  selbox[i * 5 + 3] = sb[3];
  selbox[i * 5 + 4] = sb[4];
  selcx0[i] = sb[0];                                // original cx for dets
  sellab[i] = c;
}

// ---------------- kernel 6: NMS suppression-bit tiles ----------------
// One wave per 16x16 tile of the 512x512 matrix. The whole 512-box table
// (10 KB) is DMA'd into LDS once per block by the TDM; WMMA prefilters.
__global__ __launch_bounds__(128) void roih_nms_tile_kernel(
    const float* __restrict__ selbox, unsigned short* __restrict__ supp) {
  __shared__ float sSEL[KNMS_ * 5];

  int wave = (int)((blockIdx.x * blockDim.x + threadIdx.x) >> 5);
  int lane = (int)(threadIdx.x & 31);
  const int TPB = (KNMS_ / 16) * (KNMS_ / 16);  // 1024 tiles per batch
  int b = wave / TPB;            // block-uniform: 256 blocks per batch
  int t = wave % TPB;
  int ti = t / (KNMS_ / 16);
  int tj = t % (KNMS_ / 16);

  if (threadIdx.x < 32) {
    tdm_load_1d(selbox + (size_t)b * KNMS_ * 5, lds_offset_of(sSEL),
                KNMS_ * 5);
  }
  __syncthreads();

  int j = tj * 16 + (lane & 15);
  RBox Bj; Bj.cx = sSEL[j * 5 + 0]; Bj.cy = sSEL[j * 5 + 1];
  Bj.w = sSEL[j * 5 + 2]; Bj.h = sSEL[j * 5 + 3]; Bj.a = sSEL[j * 5 + 4];
  float rj = 0.5f * sqrtf(Bj.w * Bj.w + Bj.h * Bj.h);
  v2f Bop;
  if (lane < 16) { Bop[0] = -2.0f * Bj.cx; Bop[1] = -2.0f * Bj.cy; }
  else { Bop[0] = 1.0f; Bop[1] = Bj.cx * Bj.cx + Bj.cy * Bj.cy; }

  int i0 = ti * 16 + (lane & 15);
  float ix = sSEL[i0 * 5 + 0], iy = sSEL[i0 * 5 + 1];
  v2f Aop;
  if (lane < 16) { Aop[0] = ix;            Aop[1] = iy;   }
  else           { Aop[0] = ix * ix + iy * iy; Aop[1] = 1.0f; }

  v8f acc = {};
  acc = wmma_dist(Aop, Bop, acc);   // acc[v] = d^2(row ti*16+v+8h, col j)

  int h = lane >> 4;
  unsigned short* srow = supp + (size_t)b * KNMS_ * (KNMS_ / 16);
#pragma unroll
  for (int v = 0; v < 8; ++v) {
    int i = ti * 16 + v + 8 * h;
    RBox Bi; Bi.cx = sSEL[i * 5 + 0]; Bi.cy = sSEL[i * 5 + 1];
    Bi.w = sSEL[i * 5 + 2]; Bi.h = sSEL[i * 5 + 3]; Bi.a = sSEL[i * 5 + 4];
    float ri = 0.5f * sqrtf(Bi.w * Bi.w + Bi.h * Bi.h);
    float rr = ri + rj;
    float iou = 0.0f;
    if (acc[v] < rr * rr) iou = rotated_iou_boxes(Bi, Bj);
    unsigned int bal = (unsigned int)__ballot(iou > NMS_T_);
    // lanes 0-15 of the ballot are row (ti*16+v), lanes 16-31 row (+8)
    if (lane == 0)
      srow[(size_t)(ti * 16 + v) * (KNMS_ / 16) + tj] =
          (unsigned short)(bal & 0xffffu);
    if (lane == 1)
      srow[(size_t)(ti * 16 + v + 8) * (KNMS_ / 16) + tj] =
          (unsigned short)(bal >> 16);
  }
}

// ---------------- kernel 7: greedy NMS + detection output ------------
__global__ __launch_bounds__(32) void roih_greedy_kernel(
    const unsigned short* __restrict__ supp, const float* __restrict__ tvals,
    const float* __restrict__ selbox, const float* __restrict__ selcx0,
    const int* __restrict__ sellab, float* __restrict__ dets) {
  int b = (int)blockIdx.x;
  int lane = (int)threadIdx.x;   // 32 lanes, lane owns cols [16*lane, 16*lane+15]
  unsigned int keepw = 0;
#pragma unroll
  for (int k = 0; k < 16; ++k) {
    int jj = lane * 16 + k;
    if (tvals[b * KNMS_ + jj] > 0.0f) keepw |= (1u << k);
  }
  const unsigned short* srow = supp + (size_t)b * KNMS_ * (KNMS_ / 16);
  for (int i = 0; i < KNMS_; ++i) {
    unsigned int kw = (unsigned int)__shfl((int)keepw, i >> 4, 32);
    if ((kw >> (i & 15)) & 1u) {
      unsigned int s = (unsigned int)srow[(size_t)i * (KNMS_ / 16) + lane];
      int base = lane * 16;
      unsigned int gm;
      if (base > i) gm = 0xffffu;
      else if (base + 15 <= i) gm = 0u;
      else gm = (0xffffu << (i - base + 1)) & 0xffffu;
      keepw &= ~(s & gm);
    }
  }
  __shared__ unsigned int skeep[32];
  skeep[lane] = keepw;
  __syncthreads();
  if (lane == 0) {
    float* drow = dets + (size_t)b * DETS_ * 7;
    int cnt = 0;
    for (int jj = 0; jj < KNMS_ && cnt < DETS_; ++jj) {
      if ((skeep[jj >> 4] >> (jj & 15)) & 1u) {
        const float* sb = selbox + ((size_t)b * KNMS_ + jj) * 5;
        drow[cnt * 7 + 0] = selcx0[b * KNMS_ + jj];
        drow[cnt * 7 + 1] = sb[1];
        drow[cnt * 7 + 2] = sb[2];
        drow[cnt * 7 + 3] = sb[3];
        drow[cnt * 7 + 4] = sb[4];
        drow[cnt * 7 + 5] = tvals[b * KNMS_ + jj];
        drow[cnt * 7 + 6] = (float)sellab[b * KNMS_ + jj];
        cnt++;
      }
    }
    for (; cnt < DETS_; ++cnt) {
#pragma unroll
      for (int q = 0; q < 7; ++q) drow[cnt * 7 + q] = 0.0f;
    }
  }
}

// ---------------- launch ---------------------------------------------
extern "C" void kernel_launch(void* const* d_in, const int* in_sizes, int n_in,
                              void* d_out, int out_size, void* d_ws,
                              size_t ws_size, hipStream_t stream) {
  (void)in_sizes; (void)n_in; (void)out_size; (void)ws_size;
  const float* props  = (const float*)d_in[0];  // (B,N,5)
  const float* gt     = (const float*)d_in[1];  // (B,G,5)
  const int*   gtlab  = (const int*)d_in[2];    // (B,G)
  const float* logits = (const float*)d_in[3];  // (B*N,C)
  const float* reg    = (const float*)d_in[4];  // (B*N,C*5)
  const float* rnd    = (const float*)d_in[5];  // (B,N)

  float* out = (float*)d_out;
  float* out_dets = out;                              // B*DETS*7 = 2800
  float* out_mlab = out + (size_t)B_ * DETS_ * 7;     // B*N
  float* out_mask = out_mlab + (size_t)B_ * N_;       // B*N

  // workspace carve-up (256B aligned chunks)
  char* w = (char*)d_ws;
  size_t o = 0;
  auto take = [&](size_t bytes) {
    char* p = w + o;
    o += (bytes + 255) & ~(size_t)255;
    return p;
  };
  float* pred_boxes = (float*)take((size_t)B_ * N_ * C_ * 5 * sizeof(float));
  float* cand       = (float*)take((size_t)B_ * N_ * (C_ - 1) * sizeof(float));
  float* tvals      = (float*)take((size_t)B_ * KNMS_ * sizeof(float));
  int*   tidx       = (int*)take((size_t)B_ * KNMS_ * sizeof(int));
  float* selbox     = (float*)take((size_t)B_ * KNMS_ * 5 * sizeof(float));
  float* selcx0     = (float*)take((size_t)B_ * KNMS_ * sizeof(float));
  int*   sellab     = (int*)take((size_t)B_ * KNMS_ * sizeof(int));
  unsigned short* supp =
      (unsigned short*)take((size_t)B_ * KNMS_ * (KNMS_ / 16) *
                            sizeof(unsigned short));
  int*   mlab_i     = (int*)take((size_t)B_ * N_ * sizeof(int));

  // 1. matching: B*(N/16) waves, 4 waves per 128-thread block
  {
    int waves = B_ * (N_ / 16);
    roih_match_kernel<<<waves / 4, 128, 0, stream>>>(props, gt, gtlab, mlab_i,
                                                     out_mlab);
  }
  // 2. decode + softmax + candidate scores
  {
    int threads = B_ * N_;
    roih_decode_kernel<<<(threads + 255) / 256, 256, 0, stream>>>(
        props, logits, reg, pred_boxes, cand);
  }
  // 3. subsample (needs matched labels)
  roih_subsample_kernel<<<B_, 256, 0, stream>>>(mlab_i, rnd, out_mask);
  // 4. top-512 per batch
  roih_topk_kernel<<<B_, 512, 0, stream>>>(cand, tvals, tidx);
  // 5. gather selected boxes
  {
    int threads = B_ * KNMS_;
    roih_gather_kernel<<<(threads + 255) / 256, 256, 0, stream>>>(
        pred_boxes, tidx, selbox, selcx0, sellab);
  }
  // 6. 512x512 suppression bits: B*1024 waves, 4 per block
  {
    int waves = B_ * (KNMS_ / 16) * (KNMS_ / 16);
    roih_nms_tile_kernel<<<waves / 4, 128, 0, stream>>>(selbox, supp);
  }
  // 7. greedy NMS + dets
  roih_greedy_kernel<<<B_, 32, 0, stream>>>(supp, tvals, selbox, selcx0,
                                            sellab, out_dets);
}